// GPRGNNLayer_11888469475394
// MI455X (gfx1250) — compile-verified
//
#include <hip/hip_runtime.h>
#include <hip/hip_bf16.h>

// GPR-GNN propagation for MI455X (gfx1250, wave32).
//   hidden = sum_{k=0..K} temp[k] * Ahat^k x,  Ahat = D^-1/2 (A+I) D^-1/2
// Bandwidth/atomic-bound SpMM (edge data fits in 192MB L2); dense axpy
// accumulation routed through V_WMMA_F32_16X16X4_F32 as D = (gamma*I) x Y + C.

#define D_FEAT 256

typedef __attribute__((ext_vector_type(2))) float v2f;
typedef __attribute__((ext_vector_type(8))) float v8f;

// ---------------- degree / norm ----------------

__global__ void k_fill1(float* __restrict__ p, int n) {
    int i = blockIdx.x * blockDim.x + threadIdx.x;
    if (i < n) p[i] = 1.0f;  // self-loop contributes 1 to every degree
}

__global__ void k_deg(const int* __restrict__ dst, float* __restrict__ deg, int E) {
    int i = blockIdx.x * blockDim.x + threadIdx.x;
    if (i < E) atomicAdd(&deg[dst[i]], 1.0f);  // integer-valued f32 adds: exact, order-independent
}

__global__ void k_rsqrt(float* __restrict__ p, int n) {
    int i = blockIdx.x * blockDim.x + threadIdx.x;
    if (i < n) {
        float d = p[i];
        p[i] = (d > 0.0f) ? (1.0f / sqrtf(d)) : 0.0f;
    }
}

// ---------------- dense streaming passes ----------------

// out = temp[tidx] * x   (float4 per thread)
__global__ void k_scale(const float* __restrict__ x, const float* __restrict__ temp, int tidx,
                        float* __restrict__ out, long n4) {
    long i = (long)blockIdx.x * blockDim.x + threadIdx.x;
    if (i >= n4) return;
    float g = temp[tidx];
    float4 v = ((const float4*)x)[i];
    ((float4*)out)[i] = make_float4(g * v.x, g * v.y, g * v.z, g * v.w);
}

// y = dinv[node]^2 * x   (self-loop term initializes the scatter target)
__global__ void k_selfloop_init(const float* __restrict__ x, const float* __restrict__ dinv,
                                float* __restrict__ y, long n4) {
    long i = (long)blockIdx.x * blockDim.x + threadIdx.x;
    if (i >= n4) return;
    int node = (int)(i >> 6);  // D_FEAT/4 == 64 float4 per node
    float w = dinv[node];
    w = w * w;
    float4 v = ((const float4*)x)[i];
    ((float4*)y)[i] = make_float4(w * v.x, w * v.y, w * v.z, w * v.w);
}

// ---------------- edge gather/scatter (one wave32 per edge) ----------------

__global__ void k_edge_scatter(const int* __restrict__ src, const int* __restrict__ dst,
                               const float* __restrict__ dinv,
                               const float* __restrict__ x, float* __restrict__ y, int E) {
    int wave = (int)((blockIdx.x * (unsigned)blockDim.x + threadIdx.x) >> 5);
    int lane = threadIdx.x & 31;
    if (wave >= E) return;  // wave-uniform
    int s = src[wave];
    int d = dst[wave];
    float nrm = dinv[s] * dinv[d];
    const float4* xs = (const float4*)(x + (size_t)s * D_FEAT);
    float* yd = y + (size_t)d * D_FEAT;
#pragma unroll
    for (int h = 0; h < 2; ++h) {
        int f4 = lane + h * 32;      // 64 float4 = 256 floats, fully coalesced 512B segments
        float4 v = xs[f4];
        int f = f4 * 4;
        atomicAdd(yd + f + 0, nrm * v.x);
        atomicAdd(yd + f + 1, nrm * v.y);
        atomicAdd(yd + f + 2, nrm * v.z);
        atomicAdd(yd + f + 3, nrm * v.w);
    }
}

// ---------------- hidden += gamma * Y via V_WMMA_F32_16X16X4_F32 ----------------
// One wave per 16(node) x 16(feat) tile. D = (gamma*I) x Ytile + Ctile, chained
// over four K=4 slices. Exact: each product is gamma*y or 0.
// Layouts (ISA 7.12.2, 32-bit, wave32):
//   A 16x4 : lane l, vgpr v -> A[M = l&15][K = v + 2*(l>=16)]
//   B  4x16: lane l, vgpr v -> B[K = v + 2*(l>=16)][N = l&15]
//   C/D 16x16: lane l, vgpr v -> [M = v + 8*(l>=16)][N = l&15]
__global__ void k_axpy_wmma(const float* __restrict__ y, const float* __restrict__ temp, int tidx,
                            float* __restrict__ hidden, int ntiles, int ftiles) {
    int wave = (int)((blockIdx.x * (unsigned)blockDim.x + threadIdx.x) >> 5);
    if (wave >= ntiles * ftiles) return;  // wave-uniform: EXEC stays all-ones for WMMA
    int lane = threadIdx.x & 31;
    int half = lane >> 4;
    int lm = lane & 15;
    int nb = (wave / ftiles) * 16;  // node base
    int fb = (wave % ftiles) * 16;  // feature base
    float gamma = temp[tidx];

    v8f c;
#pragma unroll
    for (int v = 0; v < 8; ++v)
        c[v] = hidden[(size_t)(nb + v + 8 * half) * D_FEAT + fb + lm];

#pragma unroll
    for (int kb = 0; kb < 4; ++kb) {
        v2f a, b;
#pragma unroll
        for (int v = 0; v < 2; ++v) {
            int kl = v + 2 * half;                       // local K within slice
            a[v] = (lm == 4 * kb + kl) ? gamma : 0.0f;   // slice of gamma*I
            b[v] = y[(size_t)(nb + 4 * kb + kl) * D_FEAT + fb + lm];
        }
        c = __builtin_amdgcn_wmma_f32_16x16x4_f32(false, a, false, b, (short)0, c, false, false);
    }

#pragma unroll
    for (int v = 0; v < 8; ++v)
        hidden[(size_t)(nb + v + 8 * half) * D_FEAT + fb + lm] = c[v];
}

// scalar tail for N % 16 nodes (N=100000 divides evenly; kept for safety)
__global__ void k_axpy_tail(const float* __restrict__ y, const float* __restrict__ temp, int tidx,
                            float* __restrict__ hidden, long start, long count) {
    long i = (long)blockIdx.x * blockDim.x + threadIdx.x;
    if (i >= count) return;
    hidden[start + i] += temp[tidx] * y[start + i];
}

// ---------------- host orchestration ----------------

extern "C" void kernel_launch(void* const* d_in, const int* in_sizes, int n_in,
                              void* d_out, int out_size, void* d_ws, size_t ws_size,
                              hipStream_t stream) {
    (void)n_in; (void)out_size; (void)ws_size;
    const float* x    = (const float*)d_in[0];
    const float* temp = (const float*)d_in[1];
    const int*   ei   = (const int*)d_in[2];   // (2, E) row-major

    const int ND = in_sizes[0];
    const int N  = ND / D_FEAT;
    const int K  = in_sizes[1] - 1;
    const int E  = in_sizes[2] / 2;
    const int* srcA = ei;
    const int* dstA = ei + (size_t)E;

    float* hidden = (float*)d_out;
    float* b0   = (float*)d_ws;                // ping
    float* b1   = b0 + (size_t)ND;             // pong
    float* dinv = b1 + (size_t)ND;             // degree -> d^-1/2

    const long n4 = (long)ND / 4;
    dim3 blk(256);
    int gElem = (int)((n4 + 255) / 256);
    int gNode = (N + 255) / 256;
    int gEdgeT = (E + 255) / 256;   // one thread per edge
    int gEdgeW = (E + 7) / 8;       // one wave per edge (8 waves/block)

    // normalization: deg = indeg + 1 (self-loop), then d^-1/2
    k_fill1<<<gNode, blk, 0, stream>>>(dinv, N);
    k_deg<<<gEdgeT, blk, 0, stream>>>(dstA, dinv, E);
    k_rsqrt<<<gNode, blk, 0, stream>>>(dinv, N);

    // hidden = temp[0] * x
    k_scale<<<gElem, blk, 0, stream>>>(x, temp, 0, hidden, n4);

    const int ftiles = D_FEAT / 16;
    const int ntiles = N / 16;
    const int gTile = (ntiles * ftiles * 32 + 255) / 256;
    const long tailStart = (long)ntiles * 16 * D_FEAT;
    const long tailCount = (long)ND - tailStart;

    const float* X = x;
    for (int k = 1; k <= K; ++k) {
        float* Y = (k & 1) ? b0 : b1;
        k_selfloop_init<<<gElem, blk, 0, stream>>>(X, dinv, Y, n4);
        k_edge_scatter<<<gEdgeW, blk, 0, stream>>>(srcA, dstA, dinv, X, Y, E);
        k_axpy_wmma<<<gTile, blk, 0, stream>>>(Y, temp, k, hidden, ntiles, ftiles);
        if (tailCount > 0) {
            k_axpy_tail<<<(int)((tailCount + 255) / 256), blk, 0, stream>>>(
                Y, temp, k, hidden, tailStart, tailCount);
        }
        X = Y;
    }
}